// TreeLSTM_29480655519921
// MI455X (gfx1250) — compile-verified
//
#include <hip/hip_runtime.h>
#include <hip/hip_bf16.h>

// MI455X / gfx1250 TreeLSTM.
// B=32, T=256, IDIM=HDIM=128.
// Strategy:
//   K1: transpose U matrices into ws (so recurrent GEMVs read coalesced / LDS-friendly).
//   K2: WMMA f32 pre-pass: Y[b,t,0:384] = x @ [Wi|Wo|Wu].T + [bi|bo|bu]  (no seq dep).
//   K3: persistent workgroup-per-batch recurrence with incremental hUf trick:
//       t = (wix+buf)*Σmask·c + Σmask·g + i*u,   g[k] = (h[k]@Uf.T)⊙c[k].

typedef float v2f __attribute__((ext_vector_type(2)));
typedef float v8f __attribute__((ext_vector_type(8)));

#define BB   32
#define TT   256
#define HH   128
#define BT   (BB * TT)        // 8192
#define WXW  384              // concat width of Wi|Wo|Wu outputs

// ws layout (floats)
#define OFF_Y    ((size_t)0)                       // BT*384
#define OFF_UT3  ((size_t)BT * WXW)                // 128*384
#define OFF_UFT  (OFF_UT3 + (size_t)HH * WXW)      // 128*128
#define OFF_G    (OFF_UFT + (size_t)HH * HH)       // BT*128

// ---------------------------------------------------------------------------
// K1: build UT3[k][r] = U_{r/128}[r%128][k]  (U0=Ui, U1=Uo, U2=Uu)
//     and UfT[k][r]   = Uf[r][k]
// ---------------------------------------------------------------------------
__global__ void prep_transpose(const float* __restrict__ Ui,
                               const float* __restrict__ Uo,
                               const float* __restrict__ Uu,
                               const float* __restrict__ Uf,
                               float* __restrict__ UT3,
                               float* __restrict__ UfT) {
  int idx = blockIdx.x * blockDim.x + threadIdx.x;   // 65536 total
  if (idx < HH * WXW) {
    int k = idx / WXW, r = idx % WXW;
    int q = r >> 7, row = r & 127;
    const float* U = (q == 0) ? Ui : ((q == 1) ? Uo : Uu);
    UT3[idx] = U[row * HH + k];
  } else {
    int j = idx - HH * WXW;                          // 16384
    int k = j / HH, r = j % HH;
    UfT[j] = Uf[r * HH + k];
  }
}

// ---------------------------------------------------------------------------
// K2: Y (8192 x 384) = X (8192 x 128) @ [Wi|Wo|Wu].T + [bi|bo|bu]
//     full-precision V_WMMA_F32_16X16X4_F32, chained over K=128 (32 steps).
//     grid: 512 M-tiles; block 256 = 8 waves; wave w owns N-tiles {w, w+8, w+16}.
// ---------------------------------------------------------------------------
__global__ __launch_bounds__(256) void wx_wmma(
    const float* __restrict__ X,
    const float* __restrict__ Wi, const float* __restrict__ bi,
    const float* __restrict__ Wo, const float* __restrict__ bo,
    const float* __restrict__ Wu, const float* __restrict__ bu,
    float* __restrict__ Y) {
  __shared__ float Xs[16 * 128];
  const int tid = threadIdx.x;
  const int mt  = blockIdx.x;
  const float* Xblk = X + (size_t)mt * 16 * HH;
  #pragma unroll
  for (int e = 0; e < 8; ++e) Xs[e * 256 + tid] = Xblk[e * 256 + tid];
  __syncthreads();

  const int lane  = tid & 31;
  const int wave  = tid >> 5;
  const int n     = lane & 15;           // N index (and A's M index)
  const int koff  = 2 * (lane >> 4);     // K sub-offset: lanes<16 -> {0,1}, >=16 -> {2,3}
  const int mbase = 8 * (lane >> 4);     // C/D row base

  #pragma unroll
  for (int t3 = 0; t3 < 3; ++t3) {
    const int nt = wave + 8 * t3;        // 0..23
    const float* W    = (nt < 8) ? Wi : ((nt < 16) ? Wo : Wu);
    const float* bias = (nt < 8) ? bi : ((nt < 16) ? bo : bu);
    const int rowbase = (nt & 7) * 16;
    const float* wrow = W + (size_t)(rowbase + n) * HH;
    const float* arow = Xs + n * HH;

    v8f c = {};
    #pragma unroll
    for (int kk = 0; kk < 32; ++kk) {
      v2f a  = *(const v2f*)(arow + 4 * kk + koff);
      v2f bq = *(const v2f*)(wrow + 4 * kk + koff);
      // D = A(16x4) * B(4x16) + C, f32 in / f32 accumulate
      c = __builtin_amdgcn_wmma_f32_16x16x4_f32(false, a, false, bq,
                                                (short)0, c, false, false);
    }
    const float bv = bias[rowbase + n];
    float* yout = Y + (size_t)(mt * 16) * WXW + nt * 16 + n;
    #pragma unroll
    for (int r = 0; r < 8; ++r)
      yout[(size_t)(mbase + r) * WXW] = c[r] + bv;
  }
}

// ---------------------------------------------------------------------------
// K3: sequential recurrence. One workgroup per batch element. 512 threads.
//     U matrices resident in dynamic LDS (256 KB) to avoid 256KB/step L2 reads.
// ---------------------------------------------------------------------------
__global__ __launch_bounds__(512) void treelstm_seq(
    const float* __restrict__ graph,   // (B,T,T)
    const float* __restrict__ Yg,      // (B*T, 384)
    const float* __restrict__ UT3g,    // (128,384)
    const float* __restrict__ UfTg,    // (128,128)
    const float* __restrict__ bui, const float* __restrict__ buo,
    const float* __restrict__ buu, const float* __restrict__ bufv,
    float* __restrict__ h_out, float* __restrict__ c_out,
    float* __restrict__ g_ws) {
  __shared__ float maskrow[TT];
  __shared__ float wxrow[WXW];
  __shared__ float hbar[HH], csum[HH], gsum[HH];
  __shared__ float part[3][4][HH];
  __shared__ float uvec[WXW];
  __shared__ float hjrow[HH], trow[HH];
  __shared__ float ufp[4][HH];
  __shared__ float biasl[4][HH];       // bui, buo, buu, buf
  extern __shared__ float dynls[];     // UT3 (128*384) then UfT (128*128)
  float* UT3 = dynls;
  float* UfT = dynls + HH * WXW;

  const int tid = threadIdx.x;
  const int b   = blockIdx.x;

  for (int i = tid; i < HH * WXW; i += 512) UT3[i] = UT3g[i];
  for (int i = tid; i < HH * HH;  i += 512) UfT[i] = UfTg[i];
  {
    int a = tid >> 7, hd = tid & 127;
    const float* src = (a == 0) ? bui : ((a == 1) ? buo : ((a == 2) ? buu : bufv));
    biasl[a][hd] = src[hd];
  }
  __syncthreads();

  const float* grow = graph + (size_t)b * TT * TT;
  const float* yrow = Yg    + (size_t)b * TT * WXW;
  float* hb = h_out + (size_t)b * TT * HH;
  float* cb = c_out + (size_t)b * TT * HH;
  float* gb = g_ws  + (size_t)b * TT * HH;

  for (int j = 0; j < TT; ++j) {
    // stage this step's mask row and precomputed x-projections
    if (tid < 256) maskrow[tid] = grow[(size_t)j * TT + tid];
    else           wxrow[tid - 256] = yrow[(size_t)j * WXW + (tid - 256)];
    if (tid < 128) wxrow[256 + tid] = yrow[(size_t)j * WXW + 256 + tid];
    __syncthreads();

    // masked gathers over history rows k < j  (h, c, g), 4-way K split
    {
      int p = tid >> 7, hd = tid & 127;
      float ah = 0.f, ac = 0.f, ag = 0.f;
      #pragma unroll 4
      for (int k = p; k < j; k += 4) {
        float m = maskrow[k];
        ah = fmaf(m, hb[(size_t)k * HH + hd], ah);
        ac = fmaf(m, cb[(size_t)k * HH + hd], ac);
        ag = fmaf(m, gb[(size_t)k * HH + hd], ag);
      }
      part[0][p][hd] = ah; part[1][p][hd] = ac; part[2][p][hd] = ag;
    }
    __syncthreads();
    if (tid < 384) {
      int a = tid >> 7, hd = tid & 127;
      float s = part[a][0][hd] + part[a][1][hd] + part[a][2][hd] + part[a][3][hd];
      if (a == 0) hbar[hd] = s; else if (a == 1) csum[hd] = s; else gsum[hd] = s;
    }
    __syncthreads();

    // uvec = [Ui|Uo|Uu] @ hbar   (384 dot products, LDS-resident weights)
    if (tid < 384) {
      float acc = 0.f;
      #pragma unroll 4
      for (int k = 0; k < HH; ++k)
        acc = fmaf(UT3[k * WXW + tid], hbar[k], acc);
      uvec[tid] = acc;
    }
    __syncthreads();

    // pointwise gates + state update
    if (tid < 128) {
      int hd = tid;
      float wix = wxrow[hd], wox = wxrow[128 + hd], wux = wxrow[256 + hd];
      float iv = wix + uvec[hd]       + biasl[0][hd];
      float ov = wox + uvec[128 + hd] + biasl[1][hd];
      float uv = wux + uvec[256 + hd] + biasl[2][hd];
      float tv = (wix + biasl[3][hd]) * csum[hd] + gsum[hd] + iv * uv;
      float hj = ov * tv;
      hb[(size_t)j * HH + hd] = hj;
      cb[(size_t)j * HH + hd] = tv;
      hjrow[hd] = hj; trow[hd] = tv;
    }
    __syncthreads();

    // g[j] = (h[j] @ Uf.T) ⊙ t   (incremental Uf transform on the new row)
    {
      int p = tid >> 7, hd = tid & 127;
      float acc = 0.f;
      #pragma unroll 4
      for (int k = 32 * p; k < 32 * p + 32; ++k)
        acc = fmaf(UfT[k * HH + hd], hjrow[k], acc);
      ufp[p][hd] = acc;
    }
    __syncthreads();
    if (tid < 128) {
      float s = ufp[0][tid] + ufp[1][tid] + ufp[2][tid] + ufp[3][tid];
      gb[(size_t)j * HH + tid] = s * trow[tid];
    }
    __threadfence_block();   // make h/c/g global writes visible to block
    __syncthreads();
  }
}

// ---------------------------------------------------------------------------
extern "C" void kernel_launch(void* const* d_in, const int* in_sizes, int n_in,
                              void* d_out, int out_size, void* d_ws, size_t ws_size,
                              hipStream_t stream) {
  const float* input = (const float*)d_in[0];
  const float* graph = (const float*)d_in[1];
  const float* Wi = (const float*)d_in[2];  const float* bi  = (const float*)d_in[3];
  const float* Wo = (const float*)d_in[4];  const float* bo  = (const float*)d_in[5];
  const float* Wu = (const float*)d_in[6];  const float* bu  = (const float*)d_in[7];
  const float* Ui = (const float*)d_in[8];  const float* bui = (const float*)d_in[9];
  const float* Uf = (const float*)d_in[10]; const float* buf = (const float*)d_in[11];
  const float* Uo = (const float*)d_in[12]; const float* buo = (const float*)d_in[13];
  const float* Uu = (const float*)d_in[14]; const float* buu = (const float*)d_in[15];

  float* ws  = (float*)d_ws;
  float* Y   = ws + OFF_Y;
  float* UT3 = ws + OFF_UT3;
  float* UfT = ws + OFF_UFT;
  float* g   = ws + OFF_G;

  float* h_out = (float*)d_out;
  float* c_out = h_out + (size_t)BT * HH;

  prep_transpose<<<256, 256, 0, stream>>>(Ui, Uo, Uu, Uf, UT3, UfT);
  wx_wmma<<<BT / 16, 256, 0, stream>>>(input, Wi, bi, Wo, bo, Wu, bu, Y);
  treelstm_seq<<<BB, 512, (HH * WXW + HH * HH) * sizeof(float), stream>>>(
      graph, Y, UT3, UfT, bui, buo, buu, buf, h_out, c_out, g);
}